// ComplEx_81449759801749
// MI455X (gfx1250) — compile-verified
//
#include <hip/hip_runtime.h>

// ComplEx rhs scoring on MI455X (gfx1250).
// scores[1024, 100000] = Q[1024, 1024] * ent_emb[100000, 1024]^T  (Q = complex product)
// HBM-bound: ~410MB E read + ~410MB C write => ~35us floor at 23.3TB/s.
// f32 WMMA 16x16x4; triple-buffered async global->LDS staging of E;
// async LDS->global b128 stores for C.

#define BATCH  1024
#define NENT   100000
#define KDIM   1024            // 2*RANK
#define RANK   512
#define NT_BLK 128             // N tile per block
#define KCHUNK 32              // K staged per LDS chunk
#define LDS_STRIDE 36          // 32 + 4 pad -> conflict-free b64 fragment reads
#define CHUNK_FLOATS (NT_BLK * LDS_STRIDE)   // 4608 floats per buffer
#define NCHUNK (KDIM / KCHUNK) // 32
#define OUT_STRIDE 65          // epilogue staging stride (64 + 1 pad)

typedef float v2f __attribute__((ext_vector_type(2)));
typedef float v8f __attribute__((ext_vector_type(8)));
typedef int   v4i __attribute__((ext_vector_type(4)));

#define AS1 __attribute__((address_space(1)))
#define AS3 __attribute__((address_space(3)))

#if __has_builtin(__builtin_amdgcn_global_load_async_to_lds_b128)
#define USE_ASYNC 1
#else
#define USE_ASYNC 0
#endif
#if __has_builtin(__builtin_amdgcn_global_store_async_from_lds_b128)
#define USE_ASYNC_ST 1
#else
#define USE_ASYNC_ST 0
#endif

#if USE_ASYNC || USE_ASYNC_ST
#if __has_builtin(__builtin_amdgcn_s_wait_asynccnt)
#define WAIT_ASYNC(n) __builtin_amdgcn_s_wait_asynccnt(n)
#else
#define WAIT_ASYNC(n) asm volatile("s_wait_asynccnt %0" :: "i"(n) : "memory")
#endif
#else
#define WAIT_ASYNC(n)
#endif

#if USE_ASYNC
// b128 DMA: global (AS1) -> LDS (AS3), no VGPR round trip, tracked by ASYNCcnt.
// Generic LDS pointer low 32 bits == LDS offset (ISA 10.2 aperture rules).
__device__ __forceinline__ void async_cp_b128(const float* g, float* l) {
  __builtin_amdgcn_global_load_async_to_lds_b128(
      (AS1 v4i*)(unsigned long long)g,
      (AS3 v4i*)(unsigned int)(unsigned long long)l,
      0, 0);
}
#endif
#if USE_ASYNC_ST
__device__ __forceinline__ void async_st_b128(float* g, const float* l) {
  __builtin_amdgcn_global_store_async_from_lds_b128(
      (AS1 v4i*)(unsigned long long)g,
      (AS3 v4i*)(unsigned int)(unsigned long long)l,
      0, 0);
}
#endif

// ---------------------------------------------------------------------------
// Kernel 1: gather + complex multiply, emitting Q in WMMA A-fragment layout.
// f32 A 16x4 layout (wave32): lane j (0-15) holds M=j, K=2h+{0,1}, h=lane>>4.
// Qswz float index = ((mt*256 + ks)*32 + lane)*2 + e
//   value = Q[mt*16 + (lane&15)][4*ks + 2*(lane>>4) + e].
// ---------------------------------------------------------------------------
__global__ __launch_bounds__(256) void complex_prep_kernel(
    const int* __restrict__ x, const float* __restrict__ ent,
    const float* __restrict__ rel, float* __restrict__ qswz) {
  int idx  = blockIdx.x * 256 + threadIdx.x;   // 0 .. 1048575
  int e    = idx & 1;
  int lane = (idx >> 1) & 31;
  int ks   = (idx >> 6) & 255;
  int mt   = idx >> 14;
  int m    = mt * 16 + (lane & 15);
  int kk   = ks * 4 + ((lane >> 4) << 1) + e;

  int eidx = x[m * 3 + 0];
  int ridx = x[m * 3 + 1];
  const float* lhs = ent + (size_t)eidx * KDIM;
  const float* rl  = rel + (size_t)ridx * KDIM;

  float q;
  if (kk < RANK) {
    float lre = lhs[kk], lim = lhs[kk + RANK];
    float rre = rl[kk],  rim = rl[kk + RANK];
    q = lre * rre - lim * rim;
  } else {
    int jj = kk - RANK;
    float lre = lhs[jj], lim = lhs[jj + RANK];
    float rre = rl[jj],  rim = rl[jj + RANK];
    q = lre * rim + lim * rre;
  }
  qswz[idx] = q;
}

// ---------------------------------------------------------------------------
// Kernel 2: C = Q * E^T via V_WMMA_F32_16X16X4_F32.
// Block: 256 threads = 8 waves; tile 128(M) x 128(N); wave w owns m-subtile w,
// 8 n-subtiles -> 8 v8f accumulators (64 VGPRs). E triple-buffered in LDS via
// async DMA; A fragments streamed from L2-resident Qswz.
// ---------------------------------------------------------------------------
__global__ __launch_bounds__(256) void complex_score_kernel(
    const float* __restrict__ E, const float* __restrict__ Qswz,
    float* __restrict__ out) {
  __shared__ float lds[3 * CHUNK_FLOATS];     // 55,296 B (epilogue reuses 33,280 B)

  const int tid  = threadIdx.x;
  const int lane = tid & 31;
  const int wave = tid >> 5;
  const int bx   = blockIdx.x;                // m-tile (fast dim -> E stripe L2 reuse)
  const int n0   = blockIdx.y * NT_BLK;
  const int mti  = bx * 8 + wave;             // 16-row m-subtile id
  const int j    = lane & 15;
  const int h    = lane >> 4;

  // Staging geometry: 4 float4 per thread per chunk (128 rows x 32 floats).
  int srow[4], scol[4];
  size_t goff[4];
#pragma unroll
  for (int i = 0; i < 4; ++i) {
    int id  = tid + 256 * i;                  // float4 id, 0..1023
    srow[i] = id >> 3;                        // 0..127
    scol[i] = (id & 7) << 2;                  // 0..28
    int gr  = n0 + srow[i]; if (gr > NENT - 1) gr = NENT - 1;
    goff[i] = (size_t)gr * KDIM + scol[i];
  }

  v8f acc[8];
  const v8f vzero = {0.f, 0.f, 0.f, 0.f, 0.f, 0.f, 0.f, 0.f};
#pragma unroll
  for (int nt = 0; nt < 8; ++nt) acc[nt] = vzero;

  const v2f* aPtr = ((const v2f*)Qswz) + (size_t)mti * (KDIM / 4) * 32 + lane;

#if USE_ASYNC
  float* bufA = lds;                          // chunk kc
  float* bufB = lds + CHUNK_FLOATS;           // chunk kc+1
  float* bufC = lds + 2 * CHUNK_FLOATS;       // chunk kc+2 (in flight)
  // prologue: chunks 0 and 1
#pragma unroll
  for (int i = 0; i < 4; ++i)
    async_cp_b128(E + goff[i], &bufA[srow[i] * LDS_STRIDE + scol[i]]);
#pragma unroll
  for (int i = 0; i < 4; ++i)
    async_cp_b128(E + goff[i] + KCHUNK, &bufB[srow[i] * LDS_STRIDE + scol[i]]);

  for (int kc = 0; kc < NCHUNK; ++kc) {
    if (kc + 2 < NCHUNK) {
      const int k0 = (kc + 2) * KCHUNK;
#pragma unroll
      for (int i = 0; i < 4; ++i)
        async_cp_b128(E + goff[i] + k0, &bufC[srow[i] * LDS_STRIDE + scol[i]]);
      WAIT_ASYNC(8);                          // chunk kc landed; kc+1/kc+2 in flight
    } else if (kc + 1 < NCHUNK) {
      WAIT_ASYNC(4);
    } else {
      WAIT_ASYNC(0);
    }
    __syncthreads();                          // everyone's DMA for kc visible
    const float* cur = bufA;
    for (int ksl = 0; ksl < KCHUNK / 4; ++ksl) {
      v2f a = aPtr[(kc * (KCHUNK / 4) + ksl) * 32];
      const int bk = ksl * 4 + 2 * h;
      v2f bf[8];
#pragma unroll
      for (int nt = 0; nt < 8; ++nt)
        bf[nt] = *(const v2f*)(cur + (nt * 16 + j) * LDS_STRIDE + bk);
#pragma unroll
      for (int nt = 0; nt < 8; ++nt)
        acc[nt] = __builtin_amdgcn_wmma_f32_16x16x4_f32(
            false, a, false, bf[nt], (short)0, acc[nt], false, false);
    }
    __syncthreads();                          // done reading before buffer reuse
    float* t = bufA; bufA = bufB; bufB = bufC; bufC = t;
  }
#else
  // ---- fallback: synchronous staging ----
  for (int kc = 0; kc < NCHUNK; ++kc) {
    const int k0 = kc * KCHUNK;
    __syncthreads();
    float4 tmp[4];
#pragma unroll
    for (int i = 0; i < 4; ++i) {
      tmp[i] = *(const float4*)(E + goff[i] + k0);
      if (kc + 1 < NCHUNK)
        __builtin_prefetch(E + goff[i] + k0 + KCHUNK, 0, 0);
    }
#pragma unroll
    for (int i = 0; i < 4; ++i)
      *(float4*)(&lds[srow[i] * LDS_STRIDE + scol[i]]) = tmp[i];
    __syncthreads();
    for (int ksl = 0; ksl < KCHUNK / 4; ++ksl) {
      v2f a = aPtr[(kc * (KCHUNK / 4) + ksl) * 32];
      const int bk = ksl * 4 + 2 * h;
      v2f bf[8];
#pragma unroll
      for (int nt = 0; nt < 8; ++nt)
        bf[nt] = *(const v2f*)(&lds[(nt * 16 + j) * LDS_STRIDE + bk]);
#pragma unroll
      for (int nt = 0; nt < 8; ++nt)
        acc[nt] = __builtin_amdgcn_wmma_f32_16x16x4_f32(
            false, a, false, bf[nt], (short)0, acc[nt], false, false);
    }
  }
#endif

  // ---- epilogue: stage C tile halves in LDS, DMA out as coalesced b128 ----
  __syncthreads();
#pragma unroll
  for (int half = 0; half < 2; ++half) {
#pragma unroll
    for (int nt2 = 0; nt2 < 4; ++nt2) {
      const int nt = half * 4 + nt2;
#pragma unroll
      for (int v = 0; v < 8; ++v) {
        lds[(wave * 16 + v + 8 * h) * OUT_STRIDE + nt2 * 16 + j] = acc[nt][v];
      }
    }
    __syncthreads();
#pragma unroll
    for (int i = 0; i < 8; ++i) {
      int id  = tid + 256 * i;                // float4 id, 0..2047
      int row = id >> 4;                      // 0..127
      int c4  = (id & 15) << 2;               // 0..60
      int col = n0 + half * 64 + c4;
      if (col < NENT) {
        float*       g = out + (size_t)(bx * 128 + row) * NENT + col;
        const float* l = &lds[row * OUT_STRIDE + c4];
#if USE_ASYNC_ST
        async_st_b128(g, l);                  // LDS -> global DMA
#else
        *(float4*)g = *(const float4*)l;
#endif
      }
    }
#if USE_ASYNC_ST
    WAIT_ASYNC(0);                            // drain before LDS restage
#endif
    __syncthreads();
  }
}

// ---------------------------------------------------------------------------
extern "C" void kernel_launch(void* const* d_in, const int* in_sizes, int n_in,
                              void* d_out, int out_size, void* d_ws, size_t ws_size,
                              hipStream_t stream) {
  const int*   x   = (const int*)d_in[0];     // [1024, 3] int32
  const float* ent = (const float*)d_in[1];   // [100000, 1024] f32
  const float* rel = (const float*)d_in[2];   // [2000, 1024] f32
  float* out  = (float*)d_out;                // [1024, 100000] f32
  float* qswz = (float*)d_ws;                 // 4 MiB scratch

  complex_prep_kernel<<<4096, 256, 0, stream>>>(x, ent, rel, qswz);

  dim3 grid(BATCH / 128, (NENT + NT_BLK - 1) / NT_BLK);  // (8, 782)
  complex_score_kernel<<<grid, 256, 0, stream>>>(ent, qswz, out);
}